// MonotonicVectorGatedChannelStack_13073880449227
// MI455X (gfx1250) — compile-verified
//
#include <hip/hip_runtime.h>
#include <math.h>

#define BDIM   4096
#define DDIM   2048
#define NCOMP  4
#define DCOMP  1024
#define OUTF   (NCOMP * DCOMP)   // 4096

typedef __attribute__((ext_vector_type(2))) float v2f;
typedef __attribute__((ext_vector_type(8))) float v8f;
typedef __attribute__((ext_vector_type(4))) int   v4i;

// ---- CDNA5 async global->LDS path (ASYNCcnt-tracked), guarded by probes ----
#if defined(__has_builtin)
#  if __has_builtin(__builtin_amdgcn_global_load_async_to_lds_b128) && \
      __has_builtin(__builtin_amdgcn_global_load_async_to_lds_b32)
#    define HAVE_ASYNC_LDS 1
#  endif
#  if __has_builtin(__builtin_amdgcn_s_wait_asynccnt)
#    define WAIT_ASYNC0() __builtin_amdgcn_s_wait_asynccnt(0)
#  else
#    define WAIT_ASYNC0() asm volatile("s_wait_asynccnt 0x0" ::: "memory")
#  endif
#endif

#ifdef HAVE_ASYNC_LDS
// builtin signature (from probe diagnostics): typed non-const pointers,
// global (AS1) source first, LDS (AS3) destination second, imm offset, cpol.
#define AS1P(T, p) ((__attribute__((address_space(1))) T*)(p))
#define AS3P(T, p) ((__attribute__((address_space(3))) T*)(p))
#define ASYNC_B128(g, l)                                                      \
    __builtin_amdgcn_global_load_async_to_lds_b128(AS1P(v4i, (g)),            \
                                                   AS3P(v4i, (l)), 0, 0)
#define ASYNC_B32(g, l)                                                       \
    __builtin_amdgcn_global_load_async_to_lds_b32(AS1P(int, (g)),             \
                                                  AS3P(int, (l)), 0, 0)
#endif

// ---------------------------------------------------------------------------
// Kernel 1: gating.  One wave (32 lanes) per row of x.
// g = x@Wg + bg, sp = softplus(x@Wn + bn), H = g + noise*sp,
// k = argmax(H), mask[i] = (i <= k).  G (== mask, straight-through forward
// value) is written into the G region of d_out; kernel 2 re-reads it.
// ---------------------------------------------------------------------------
__global__ void __launch_bounds__(256)
gate_kernel(const float* __restrict__ x, const float* __restrict__ noise,
            const float* __restrict__ Wg, const float* __restrict__ bg,
            const float* __restrict__ Wn, const float* __restrict__ bn,
            float* __restrict__ Gout)
{
    const int lane = threadIdx.x & 31;
    const int wid  = threadIdx.x >> 5;
    const int row  = blockIdx.x * 8 + wid;

    const float* xr = x + (size_t)row * DDIM;
    const float4* wg4 = (const float4*)Wg;   // Wg is [D, 4] -> one float4 per d
    const float4* wn4 = (const float4*)Wn;

    float g0 = 0.f, g1 = 0.f, g2 = 0.f, g3 = 0.f;
    float s0 = 0.f, s1 = 0.f, s2 = 0.f, s3 = 0.f;

    for (int d = lane; d < DDIM; d += 32) {
        float  xv = xr[d];
        float4 wg = wg4[d];
        float4 wn = wn4[d];
        g0 = fmaf(xv, wg.x, g0); g1 = fmaf(xv, wg.y, g1);
        g2 = fmaf(xv, wg.z, g2); g3 = fmaf(xv, wg.w, g3);
        s0 = fmaf(xv, wn.x, s0); s1 = fmaf(xv, wn.y, s1);
        s2 = fmaf(xv, wn.z, s2); s3 = fmaf(xv, wn.w, s3);
    }
    #pragma unroll
    for (int off = 16; off >= 1; off >>= 1) {
        g0 += __shfl_xor(g0, off, 32); g1 += __shfl_xor(g1, off, 32);
        g2 += __shfl_xor(g2, off, 32); g3 += __shfl_xor(g3, off, 32);
        s0 += __shfl_xor(s0, off, 32); s1 += __shfl_xor(s1, off, 32);
        s2 += __shfl_xor(s2, off, 32); s3 += __shfl_xor(s3, off, 32);
    }

    if (lane == 0) {
        float gv[4] = {g0, g1, g2, g3};
        float sv[4] = {s0, s1, s2, s3};
        float H[4];
        #pragma unroll
        for (int n = 0; n < NCOMP; ++n) {
            float z  = sv[n] + bn[n];
            float sp = (z > 20.0f) ? z : log1pf(__expf(z));   // softplus
            H[n] = gv[n] + bg[n] + noise[row * NCOMP + n] * sp;
        }
        int   k    = 0;
        float best = H[0];
        #pragma unroll
        for (int n = 1; n < NCOMP; ++n)
            if (H[n] > best) { best = H[n]; k = n; }          // first-max tie rule
        #pragma unroll
        for (int n = 0; n < NCOMP; ++n)
            Gout[row * NCOMP + n] = (n <= k) ? 1.0f : 0.0f;
    }
}

// ---------------------------------------------------------------------------
// Kernel 2: masked GEMM
//   E[b, n*1024+o] = mask[b,n] * (sum_d x[b,d]*Wc[n,d,o] + bc[n,o])
// Block tile 128x128, double-buffered BK=16 stages, 8 waves (4 M x 2 N),
// each wave owns 2x4 subtiles of 16x16 via V_WMMA_F32_16X16X4_F32.
// LDS layouts (stride 20, even & =0 mod 4):
//   sA[row][k]  -> A fragment (k,k+1) is a contiguous, 8B-aligned ds_load_b64
//   sB[col][k]  -> B stored TRANSPOSED so B fragment is likewise one b64 load
// Buffer fill uses GLOBAL_LOAD_ASYNC_TO_LDS (ASYNCcnt) when available: zero
// staging VGPRs, fill fully overlapped with WMMA compute, no spills.
// ---------------------------------------------------------------------------
#define BM 128
#define BN 128
#define BK 16
#define LDT 20            // LDS stride (dwords) for both sA rows and sB cols

__global__ void __launch_bounds__(256)
gemm_mask_kernel(const float* __restrict__ x, const float* __restrict__ Wc,
                 const float* __restrict__ bc, const float* __restrict__ gmask,
                 float* __restrict__ E)
{
    __shared__ float sA[2][BM * LDT];   // 2 x 10240 B
    __shared__ float sB[2][BN * LDT];   // 2 x 10240 B
    __shared__ float sMask[BM];

    const int tid  = threadIdx.x;
    const int lane = tid & 31;
    const int wid  = tid >> 5;

    const int m0 = blockIdx.y * BM;
    const int n0 = blockIdx.x * BN;
    const int ncomp = n0 / DCOMP;       // 128-col tile lies inside one component
    const int ocol0 = n0 % DCOMP;

    const int wm = wid & 3;             // wave row group: 32 rows each
    const int wn = wid >> 2;            // wave col group: 64 cols each

    v8f acc[2][4] = {};

    if (tid < BM) sMask[tid] = gmask[(size_t)(m0 + tid) * NCOMP + ncomp];

    const float* xblk = x + (size_t)m0 * DDIM;
    const float* wblk = Wc + (size_t)ncomp * DDIM * DCOMP + ocol0;

    // staging assignment
    //   A: 128x16 f32 = 512 float4; thread covers float4 idx {tid, tid+256}
    const int ar0 = tid >> 2;           // row 0..63 (first half)
    const int ac4 = tid & 3;            // float4 col 0..3
    //   B: 16x128 f32; thread owns one column, 8 consecutive k rows
    const int bcol = tid & 127;
    const int brow = (tid >> 7) * 8;    // 0 or 8

    const int lm    = lane & 15;
    const int khalf = (lane >> 4) << 1;  // lanes 0-15 -> K+0/1, lanes 16-31 -> K+2/3

#ifdef HAVE_ASYNC_LDS
    // ---------------- async global->LDS double-buffer pipeline -------------
    #define ISSUE_TILE(k0, buf)                                               \
        do {                                                                  \
            const float* ap_ = xblk + (k0);                                   \
            ASYNC_B128(ap_ + (size_t)ar0        * DDIM + ac4 * 4,             \
                       &sA[(buf)][ar0 * LDT + ac4 * 4]);                      \
            ASYNC_B128(ap_ + (size_t)(ar0 + 64) * DDIM + ac4 * 4,             \
                       &sA[(buf)][(ar0 + 64) * LDT + ac4 * 4]);               \
            const float* bp_ = wblk + (size_t)(k0) * DCOMP;                   \
            _Pragma("unroll")                                                 \
            for (int i_ = 0; i_ < 8; ++i_)                                    \
                ASYNC_B32(bp_ + (size_t)(brow + i_) * DCOMP + bcol,           \
                          &sB[(buf)][bcol * LDT + brow + i_]);                \
        } while (0)

    ISSUE_TILE(0, 0);

    int cur = 0;
    for (int k0 = 0; k0 < DDIM; k0 += BK) {
        WAIT_ASYNC0();       // this thread's tile-k0 LDS writes are complete
        __syncthreads();     // all writes visible; other buffer free for fill

        const bool more = (k0 + BK) < DDIM;
        if (more) {
            ISSUE_TILE(k0 + BK, cur ^ 1);   // in flight across the compute
            if (k0 + 2 * BK < DDIM)         // warm L2 for the tile after next
                __builtin_prefetch(xblk + (size_t)ar0 * DDIM + k0 + 2 * BK, 0, 1);
        }

        const float* __restrict__ As = sA[cur];
        const float* __restrict__ Bs = sB[cur];
        #pragma unroll
        for (int kk = 0; kk < BK; kk += 4) {
            v2f afr[2], bfr[4];
            #pragma unroll
            for (int i = 0; i < 2; ++i)
                afr[i] = *(const v2f*)&As[(wm * 32 + i * 16 + lm) * LDT + kk + khalf];
            #pragma unroll
            for (int j = 0; j < 4; ++j)
                bfr[j] = *(const v2f*)&Bs[(wn * 64 + j * 16 + lm) * LDT + kk + khalf];
            #pragma unroll
            for (int i = 0; i < 2; ++i)
                #pragma unroll
                for (int j = 0; j < 4; ++j)
                    acc[i][j] = __builtin_amdgcn_wmma_f32_16x16x4_f32(
                        false, afr[i], false, bfr[j],
                        (short)0, acc[i][j], false, false);
        }
        cur ^= 1;
    }
    #undef ISSUE_TILE
#else
    // ---------------- fallback: register-staged double buffering -----------
    float4 aReg[2];
    float  bReg[8];

    aReg[0] = *(const float4*)(xblk + (size_t)ar0        * DDIM + ac4 * 4);
    aReg[1] = *(const float4*)(xblk + (size_t)(ar0 + 64) * DDIM + ac4 * 4);
    #pragma unroll
    for (int i = 0; i < 8; ++i)
        bReg[i] = wblk[(size_t)(brow + i) * DCOMP + bcol];

    *(float4*)&sA[0][ar0 * LDT + ac4 * 4]        = aReg[0];
    *(float4*)&sA[0][(ar0 + 64) * LDT + ac4 * 4] = aReg[1];
    *(float4*)&sB[0][bcol * LDT + brow]     = make_float4(bReg[0], bReg[1], bReg[2], bReg[3]);
    *(float4*)&sB[0][bcol * LDT + brow + 4] = make_float4(bReg[4], bReg[5], bReg[6], bReg[7]);

    int cur = 0;
    for (int k0 = 0; k0 < DDIM; k0 += BK) {
        const bool more = (k0 + BK) < DDIM;
        if (more) {
            const float* ap = xblk + k0 + BK;
            aReg[0] = *(const float4*)(ap + (size_t)ar0        * DDIM + ac4 * 4);
            aReg[1] = *(const float4*)(ap + (size_t)(ar0 + 64) * DDIM + ac4 * 4);
            const float* bp = wblk + (size_t)(k0 + BK) * DCOMP;
            #pragma unroll
            for (int i = 0; i < 8; ++i)
                bReg[i] = bp[(size_t)(brow + i) * DCOMP + bcol];
        }

        __syncthreads();

        const float* __restrict__ As = sA[cur];
        const float* __restrict__ Bs = sB[cur];
        #pragma unroll
        for (int kk = 0; kk < BK; kk += 4) {
            v2f afr[2], bfr[4];
            #pragma unroll
            for (int i = 0; i < 2; ++i)
                afr[i] = *(const v2f*)&As[(wm * 32 + i * 16 + lm) * LDT + kk + khalf];
            #pragma unroll
            for (int j = 0; j < 4; ++j)
                bfr[j] = *(const v2f*)&Bs[(wn * 64 + j * 16 + lm) * LDT + kk + khalf];
            #pragma unroll
            for (int i = 0; i < 2; ++i)
                #pragma unroll
                for (int j = 0; j < 4; ++j)
                    acc[i][j] = __builtin_amdgcn_wmma_f32_16x16x4_f32(
                        false, afr[i], false, bfr[j],
                        (short)0, acc[i][j], false, false);
        }

        if (more) {
            float* An = sA[cur ^ 1];
            float* Bn = sB[cur ^ 1];
            *(float4*)&An[ar0 * LDT + ac4 * 4]        = aReg[0];
            *(float4*)&An[(ar0 + 64) * LDT + ac4 * 4] = aReg[1];
            *(float4*)&Bn[bcol * LDT + brow]     = make_float4(bReg[0], bReg[1], bReg[2], bReg[3]);
            *(float4*)&Bn[bcol * LDT + brow + 4] = make_float4(bReg[4], bReg[5], bReg[6], bReg[7]);
        }
        cur ^= 1;
    }
#endif

    // ---- epilogue: +bias, *mask, store ----
    const int mhalf = (lane >> 4) << 3;  // lanes 0-15 -> M 0..7, lanes 16-31 -> M 8..15
    #pragma unroll
    for (int i = 0; i < 2; ++i) {
        #pragma unroll
        for (int j = 0; j < 4; ++j) {
            int colLocal = wn * 64 + j * 16 + lm;
            int col      = n0 + colLocal;
            float bias   = bc[ncomp * DCOMP + ocol0 + colLocal];
            #pragma unroll
            for (int r = 0; r < 8; ++r) {
                int rlocal = wm * 32 + i * 16 + mhalf + r;
                int row    = m0 + rlocal;
                float v    = (acc[i][j][r] + bias) * sMask[rlocal];
                E[(size_t)row * OUTF + col] = v;
            }
        }
    }
}

// ---------------------------------------------------------------------------
// Launch: inputs in setup_inputs() order:
//   0:x  1:noise  2:Wc  3:bc  4:Wg  5:bg  6:Wn  7:bn
// d_out = [ E (4096*4096 f32) | G (4096*4 f32) ]
// ---------------------------------------------------------------------------
extern "C" void kernel_launch(void* const* d_in, const int* in_sizes, int n_in,
                              void* d_out, int out_size, void* d_ws, size_t ws_size,
                              hipStream_t stream)
{
    const float* x     = (const float*)d_in[0];
    const float* noise = (const float*)d_in[1];
    const float* Wc    = (const float*)d_in[2];
    const float* bc    = (const float*)d_in[3];
    const float* Wg    = (const float*)d_in[4];
    const float* bg    = (const float*)d_in[5];
    const float* Wn    = (const float*)d_in[6];
    const float* bn    = (const float*)d_in[7];

    float* E = (float*)d_out;
    float* G = E + (size_t)BDIM * OUTF;

    // 1) gating -> writes G (== mask) region of d_out
    gate_kernel<<<BDIM / 8, 256, 0, stream>>>(x, noise, Wg, bg, Wn, bn, G);

    // 2) masked WMMA GEMM -> writes E, reading mask from G
    dim3 grid(OUTF / BN, BDIM / BM);   // (32, 32)
    gemm_mask_kernel<<<grid, 256, 0, stream>>>(x, Wc, bc, G, E);
}